// Cace_19507741458666
// MI455X (gfx1250) — compile-verified
//
#include <hip/hip_runtime.h>
#include <utility>

// ---------------------------------------------------------------------------
// Problem constants (must match reference)
// ---------------------------------------------------------------------------
#define N_NODES 5000
#define N_EDGES 50000
#define N_RBF   8
#define NANG    20      // angular terms for MAX_L=3
#define NCH     9       // encoded channels (3x3)
#define NFEAT   11      // 1 + 3 + 2 + 5 output features
#define CUTOFF  5.5f
#define EPSV    1e-9f
#define PI_F    3.14159265358979323846f

// A tensor: [N_NODES][8][20][9] f32 = 7,200,000 floats = 28.8 MB (fits in L2)
#define A_ELEMS (N_NODES * N_RBF * NANG * NCH)
#define PIX     (N_NODES * N_RBF * NCH)      // 360,000 contraction "pixels"
#define NTERM   400                          // 398 real terms + 2 zero pads
#define NWMMA   (NTERM / 4)                  // 100 chained 16x16x4 WMMAs

typedef float v2f __attribute__((ext_vector_type(2)));
typedef float v8f __attribute__((ext_vector_type(8)));

// ---------------------------------------------------------------------------
// Compile-time combo-table generation (mirrors reference _build_combos)
// ---------------------------------------------------------------------------
struct LV { int x, y, z; };
constexpr LV ANG[NANG] = {
  {0,0,0},
  {1,0,0},{0,1,0},{0,0,1},
  {2,0,0},{1,1,0},{1,0,1},{0,2,0},{0,1,1},{0,0,2},
  {3,0,0},{2,1,0},{2,0,1},{1,2,0},{1,1,1},{1,0,2},{0,3,0},{0,2,1},{0,1,2},{0,0,3}
};
constexpr int angL(int i) { return ANG[i].x + ANG[i].y + ANG[i].z; }
constexpr int angIdx(int x, int y, int z) {
  for (int i = 0; i < NANG; ++i)
    if (ANG[i].x == x && ANG[i].y == y && ANG[i].z == z) return i;
  return 0;
}
constexpr double factd(int n) { double r = 1; for (int i = 2; i <= n; ++i) r *= i; return r; }
constexpr double mnom(int i) {
  return factd(angL(i)) / (factd(ANG[i].x) * factd(ANG[i].y) * factd(ANG[i].z));
}

struct Term { int i0, i1, i2, i3, seg; float pref; };
struct TermTable { Term t[NTERM]; };

constexpr TermTable buildTerms() {
  TermTable T{};
  int n = 0;
  // feature 0: identity A[..., k=0, ...]
  T.t[n++] = Term{0, -1, -1, -1, 0, 1.0f};
  int seg = 1;
  // COMBOS[0]: l = 1..3, squared terms            -> features 1..3
  for (int l = 1; l <= 3; ++l) {
    for (int i = 1; i < NANG; ++i)
      if (angL(i) == l)
        T.t[n++] = Term{i, i, -1, -1, seg, (float)mnom(i)};
    ++seg;
  }
  // COMBOS[1]: sorted (l1<=l2<=l3) triple combos  -> features 4..5
  for (int l1 = 1; l1 <= 3; ++l1)
    for (int l2 = l1; l2 <= 3; ++l2)
      for (int l3 = l2; l3 <= 3; ++l3) {
        if (l2 + l3 > 3 || l1 + l3 > 3 || l1 + l2 > 3) continue;
        for (int a = 0; a < NANG; ++a) { if (angL(a) != l1) continue;
          for (int b = 0; b < NANG; ++b) { if (angL(b) != l2) continue;
            for (int c = 0; c < NANG; ++c) { if (angL(c) != l3) continue;
              int vi = angIdx(ANG[c].x + ANG[b].x, ANG[c].y + ANG[b].y, ANG[c].z + ANG[b].z);
              int vj = angIdx(ANG[c].x + ANG[a].x, ANG[c].y + ANG[a].y, ANG[c].z + ANG[a].z);
              int vk = angIdx(ANG[b].x + ANG[a].x, ANG[b].y + ANG[a].y, ANG[b].z + ANG[a].z);
              T.t[n++] = Term{vi, vj, vk, -1, seg, (float)(mnom(a) * mnom(b) * mnom(c))};
            } } }
        ++seg;
      }
  // COMBOS[2]: chain combos (l1,l2,l3 free)       -> features 6..10
  for (int l1 = 1; l1 <= 3; ++l1)
    for (int l2 = 1; l2 <= 3; ++l2)
      for (int l3 = 1; l3 <= 3; ++l3) {
        if (l1 + l2 > 3 || l2 + l3 > 3) continue;
        for (int a = 0; a < NANG; ++a) { if (angL(a) != l1) continue;
          for (int b = 0; b < NANG; ++b) { if (angL(b) != l2) continue;
            for (int c = 0; c < NANG; ++c) { if (angL(c) != l3) continue;
              int v1 = a;
              int v2 = angIdx(ANG[a].x + ANG[b].x, ANG[a].y + ANG[b].y, ANG[a].z + ANG[b].z);
              int v3 = angIdx(ANG[b].x + ANG[c].x, ANG[b].y + ANG[c].y, ANG[b].z + ANG[c].z);
              int v4 = c;
              T.t[n++] = Term{v1, v2, v3, v4, seg, (float)(mnom(a) * mnom(b) * mnom(c))};
            } } }
        ++seg;
      }
  // pad to 400 with zero-prefactor terms (contribute 0 to feature 0)
  for (; n < NTERM; ++n) T.t[n] = Term{0, -1, -1, -1, 0, 0.0f};
  return T;
}
constexpr TermTable TT = buildTerms();   // 398 real terms, 11 segments

// ---------------------------------------------------------------------------
// Kernel 1: zero the accumulator A in workspace
// ---------------------------------------------------------------------------
__global__ void cace_zero(float4* __restrict__ a4, int n4) {
  int i = blockIdx.x * blockDim.x + threadIdx.x;
  if (i < n4) a4[i] = make_float4(0.f, 0.f, 0.f, 0.f);
}

// ---------------------------------------------------------------------------
// Kernel 2: per-edge rank-1 tensor, scattered into A with f32 atomics.
// One wave32 per edge; the 37 per-edge scalars go through LDS so the
// 45-element per-lane loop can index them dynamically (ds_load, no scratch).
// ---------------------------------------------------------------------------
__global__ void __launch_bounds__(128)
cace_edge(const float* __restrict__ pos, const float* __restrict__ shifts,
          const float* __restrict__ Wemb, const int* __restrict__ zno,
          const int* __restrict__ eidx, float* __restrict__ A) {
  const int wave = (blockIdx.x * blockDim.x + threadIdx.x) >> 5;   // edge id
  const int lane = threadIdx.x & 31;
  const int w    = (threadIdx.x >> 5);                              // wave in block

  __shared__ float sh[4][38];   // [0..7]=radial, [8..27]=ang, [28..36]=enc

  const int s   = eidx[wave];
  const int rcv = eidx[N_EDGES + wave];

  // atomic number -> embedding row (ZS = {1,6,7,8})
  const int zs = zno[s], zr = zno[rcv];
  const int rs = (zs == 1) ? 0 : (zs == 6) ? 1 : (zs == 7) ? 2 : 3;
  const int rr = (zr == 1) ? 0 : (zr == 6) ? 1 : (zr == 7) ? 2 : 3;
  const float e0 = Wemb[rs * 3 + 0], e1 = Wemb[rs * 3 + 1], e2 = Wemb[rs * 3 + 2];
  const float g0 = Wemb[rr * 3 + 0], g1 = Wemb[rr * 3 + 1], g2 = Wemb[rr * 3 + 2];

  const float vx = pos[rcv * 3 + 0] - pos[s * 3 + 0] + shifts[wave * 3 + 0];
  const float vy = pos[rcv * 3 + 1] - pos[s * 3 + 1] + shifts[wave * 3 + 1];
  const float vz = pos[rcv * 3 + 2] - pos[s * 3 + 2] + shifts[wave * 3 + 2];
  const float L   = sqrtf(vx * vx + vy * vy + vz * vz);
  const float inv = 1.0f / (L + EPSV);
  const float x = vx * inv, y = vy * inv, z = vz * inv;

  // polynomial cutoff
  const float u  = L / CUTOFF;
  const float u2 = u * u, u4 = u2 * u2, u6 = u4 * u2, u7 = u6 * u, u8 = u4 * u4;
  float fc = 1.0f - 28.0f * u6 + 48.0f * u7 - 21.0f * u8;
  fc = (L < CUTOFF) ? fc : 0.0f;
  const float rpref = sqrtf(2.0f / CUTOFF) * inv * fc;
  const float kL = PI_F * L / CUTOFF;

  if (lane == 0) {
    #pragma unroll
    for (int ri = 0; ri < N_RBF; ++ri)
      sh[w][ri] = rpref * __sinf((float)(ri + 1) * kL);
    const float xx = x * x, yy = y * y, zz = z * z;
    sh[w][8]  = 1.0f;
    sh[w][9]  = x;       sh[w][10] = y;       sh[w][11] = z;
    sh[w][12] = xx;      sh[w][13] = x * y;   sh[w][14] = x * z;
    sh[w][15] = yy;      sh[w][16] = y * z;   sh[w][17] = zz;
    sh[w][18] = xx * x;  sh[w][19] = xx * y;  sh[w][20] = xx * z;
    sh[w][21] = x * yy;  sh[w][22] = x * y * z; sh[w][23] = x * zz;
    sh[w][24] = yy * y;  sh[w][25] = yy * z;  sh[w][26] = y * zz;  sh[w][27] = zz * z;
    sh[w][28] = e0 * g0; sh[w][29] = e0 * g1; sh[w][30] = e0 * g2;
    sh[w][31] = e1 * g0; sh[w][32] = e1 * g1; sh[w][33] = e1 * g2;
    sh[w][34] = e2 * g0; sh[w][35] = e2 * g1; sh[w][36] = e2 * g2;
  }
  __syncthreads();

  float* dst = A + (long)rcv * (N_RBF * NANG * NCH);
  #pragma unroll
  for (int i = 0; i < 45; ++i) {                 // 45 * 32 = 1440 = 8*20*9
    const int flat = lane + 32 * i;
    const int r   = flat / (NANG * NCH);
    const int rem = flat - r * (NANG * NCH);
    const int k   = rem / NCH;
    const int c   = rem - k * NCH;
    const float val = sh[w][r] * sh[w][8 + k] * sh[w][28 + c];
    atomicAdd(&dst[flat], val);                  // coalesced lanes -> L2 atomics
  }
}

// ---------------------------------------------------------------------------
// Kernel 3: contraction A -> out via chained V_WMMA_F32_16X16X4_F32.
// Each wave handles 16 pixels (rows M); K = 400 padded terms, 100 WMMAs.
// All combo indices are constexpr -> pure register multiplies feed the WMMA.
// ---------------------------------------------------------------------------
template <int T0>
__device__ __forceinline__ float prodTerm(const float Av[NANG]) {
  constexpr Term t = TT.t[T0];
  if constexpr (t.pref == 0.0f) {
    return 0.0f;
  } else {
    float v = t.pref * Av[t.i0];
    if constexpr (t.i1 >= 0) v *= Av[t.i1];
    if constexpr (t.i2 >= 0) v *= Av[t.i2];
    if constexpr (t.i3 >= 0) v *= Av[t.i3];
    return v;
  }
}

template <int KT>
__device__ __forceinline__ v8f wstep(const float Av[NANG], bool hiHalf, int f, v8f acc) {
  // A 16x4 f32 layout: lanes 0-15 provide K = 4KT+{0,1}; lanes 16-31 K = 4KT+{2,3}
  const float axLo = prodTerm<4 * KT + 0>(Av);
  const float ayLo = prodTerm<4 * KT + 1>(Av);
  const float axHi = prodTerm<4 * KT + 2>(Av);
  const float ayHi = prodTerm<4 * KT + 3>(Av);
  v2f a, b;
  a.x = hiHalf ? axHi : axLo;
  a.y = hiHalf ? ayHi : ayLo;
  // B 4x16: selection matrix B[k, f] = (seg[k] == f)
  constexpr int s0 = TT.t[4 * KT + 0].seg, s1 = TT.t[4 * KT + 1].seg;
  constexpr int s2 = TT.t[4 * KT + 2].seg, s3 = TT.t[4 * KT + 3].seg;
  const int sx = hiHalf ? s2 : s0;
  const int sy = hiHalf ? s3 : s1;
  b.x = (f == sx) ? 1.0f : 0.0f;
  b.y = (f == sy) ? 1.0f : 0.0f;
  return __builtin_amdgcn_wmma_f32_16x16x4_f32(
      /*neg_a=*/false, a, /*neg_b=*/false, b,
      /*c_mod=*/(short)0, acc, /*reuse_a=*/false, /*reuse_b=*/false);
}

template <int... KT>
__device__ __forceinline__ void allSteps(std::integer_sequence<int, KT...>,
                                         const float Av[NANG], bool hiHalf, int f, v8f& acc) {
  ((acc = wstep<KT>(Av, hiHalf, f, acc)), ...);
}

__global__ void __launch_bounds__(128)
cace_contract(const float* __restrict__ A, float* __restrict__ out) {
  const int wave = (blockIdx.x * blockDim.x + threadIdx.x) >> 5;
  const int lane = threadIdx.x & 31;
  const int f    = lane & 15;            // output feature column (N dim)
  const bool hiHalf = lane >= 16;

  // pixel p = ((n*8)+r)*9 + c ; each half-wave loads the same 16 pixels
  const int p = wave * 16 + (lane & 15);
  const int n   = p / (N_RBF * NCH);
  const int rem = p - n * (N_RBF * NCH);
  const int r   = rem / NCH;
  const int c   = rem - r * NCH;

  const float* Abase = A + ((long)(n * N_RBF + r)) * (NANG * NCH) + c;
  float Av[NANG];
  #pragma unroll
  for (int k = 0; k < NANG; ++k) Av[k] = Abase[k * NCH];

  v8f acc = {};
  allSteps(std::make_integer_sequence<int, NWMMA>{}, Av, hiHalf, f, acc);

  // D 16x16 layout: VGPR j holds row M=j (lanes 0-15) / M=j+8 (lanes 16-31)
  if (f < NFEAT) {
    const int prowBase = wave * 16 + (hiHalf ? 8 : 0);
    #pragma unroll
    for (int j = 0; j < 8; ++j) {
      const int pr  = prowBase + j;
      const int nn  = pr / (N_RBF * NCH);
      const int rm  = pr - nn * (N_RBF * NCH);
      const int rr2 = rm / NCH;
      const int cc  = rm - rr2 * NCH;
      out[(((long)nn * N_RBF + rr2) * NFEAT + f) * NCH + cc] = acc[j];
    }
  }
}

// ---------------------------------------------------------------------------
// Launcher
// ---------------------------------------------------------------------------
extern "C" void kernel_launch(void* const* d_in, const int* in_sizes, int n_in,
                              void* d_out, int out_size, void* d_ws, size_t ws_size,
                              hipStream_t stream) {
  const float* positions = (const float*)d_in[0];
  const float* shifts    = (const float*)d_in[1];
  const float* W_embed   = (const float*)d_in[2];
  const int*   atomic_z  = (const int*)d_in[3];
  const int*   edge_idx  = (const int*)d_in[4];
  float* out = (float*)d_out;
  float* A   = (float*)d_ws;                  // 28.8 MB accumulator

  // 1) zero A
  const int n4 = A_ELEMS / 4;                 // 1,800,000 float4
  cace_zero<<<(n4 + 255) / 256, 256, 0, stream>>>((float4*)A, n4);

  // 2) edge scatter: one wave per edge, 4 waves/block -> 12500 blocks
  cace_edge<<<N_EDGES / 4, 128, 0, stream>>>(positions, shifts, W_embed,
                                             atomic_z, edge_idx, A);

  // 3) WMMA contraction: 22500 waves (16 pixels each), 4 waves/block
  cace_contract<<<(PIX / 16) / 4, 128, 0, stream>>>(A, out);
}